// CurveThetaMultiResGrid_71846212927744
// MI455X (gfx1250) — compile-verified
//
#include <hip/hip_runtime.h>
#include <math.h>

// CurveThetaMultiResGrid: random bilinear gather over 4 mip levels, concat.
// B*N = 262144 samples, C = 32 channels/level, out = (B*N, 128) f32.
// Latency-bound random gather; dataset (170MB) fits MI455X 192MB L2.
//
// Thread mapping (wave32):
//   wave w handles 32 samples at ONE level: level = w & 3 (wave-uniform,
//   scalarized -> grid base ptr / W / H live in SGPRs, loads use GVS
//   saddr+voffset addressing), sample = (w>>2)*32 + lane.
// Each thread: 128 independent global_load_b32 gathers, issued in batches of
// 64 (LOADcnt cap is 63) for maximum memory-level parallelism, then 8
// coalesced global_store_b128 stores.

#define PI_F 3.14159265358979323846f

__global__ __launch_bounds__(256)
void curvegrid_kernel(const float* __restrict__ ts,
                      const float* __restrict__ theta,
                      const float* __restrict__ g0,
                      const float* __restrict__ g1,
                      const float* __restrict__ g2,
                      const float* __restrict__ g3,
                      float* __restrict__ out,
                      int S)
{
    const int tid  = blockIdx.x * blockDim.x + threadIdx.x;
    const int lane = tid & 31;
    const int wave = tid >> 5;

    // level is uniform across the wave; force it scalar so the grid pointer,
    // W, H and HW end up in SGPRs (uniform branches, GVS addressing).
    const int level  = __builtin_amdgcn_readfirstlane(wave & 3);
    const int sample = ((wave >> 2) << 5) + lane;
    if (sample >= S) return;

    const int Wd = 256 << level;
    const int Hd = 64  << level;
    const int HW = Hd * Wd;                       // <= 1,048,576 (level 3)
    const float* __restrict__ g =
        (level == 0) ? g0 : (level == 1) ? g1 : (level == 2) ? g2 : g3;

    // ---- grid coords (matches reference math in fp32) ----
    const float th  = theta[sample];
    const float tsv = ts[sample];

    float t    = (th + PI_F) * (1.0f / (2.0f * PI_F));
    float wrap = t - floorf(t);                   // [0,1)
    float gx   = 2.0f * wrap - 1.0f;              // [-1,1)
    // (TS_MAX-TS_MIN+1e-12) evaluates to 1.0f in fp32, as in the reference
    float tsn  = tsv;
    float gy   = fminf(fmaxf(tsn, -1.0f), 1.0f);

    float x = (gx + 1.0f) * 0.5f * (float)(Wd - 1);
    float y = (gy + 1.0f) * 0.5f * (float)(Hd - 1);

    float x0f = floorf(x);
    float y0f = floorf(y);
    float wx  = x - x0f;
    float wy  = y - y0f;

    int x0 = (int)x0f; x0 = (x0 < 0) ? 0 : ((x0 > Wd - 1) ? Wd - 1 : x0);
    int y0 = (int)y0f; y0 = (y0 < 0) ? 0 : ((y0 > Hd - 1) ? Hd - 1 : y0);
    int x1 = (x0 + 1 > Wd - 1) ? Wd - 1 : x0 + 1;
    int y1 = (y0 + 1 > Hd - 1) ? Hd - 1 : y0 + 1;

    const float w11 = wx * wy;
    const float w10 = wy - w11;                   // (1-wx)*wy
    const float w01 = wx - w11;                   // wx*(1-wy)
    const float w00 = 1.0f - wx - wy + w11;       // (1-wx)*(1-wy)

    // 32-bit element offsets (max byte offset 134MB < 2^31)
    const int i00 = y0 * Wd + x0;
    const int i01 = y0 * Wd + x1;
    const int i10 = y1 * Wd + x0;
    const int i11 = y1 * Wd + x1;

    // Output slice for this (sample, level): 32 contiguous floats, 16B-aligned.
    float* __restrict__ op = out + (size_t)sample * 128 + level * 32;

    // 8 groups x 4 channels. Unroll 4 groups -> 64 gathers issued back-to-back
    // (right at the 6-bit LOADcnt ceiling) before the first consuming wait,
    // then 4 batched b128 stores.
#pragma unroll 4
    for (int cg = 0; cg < 8; ++cg) {
        float v[4];
#pragma unroll
        for (int j = 0; j < 4; ++j) {
            const int o   = (cg * 4 + j) * HW;    // scalar (c * HW)
            const float a = g[o + i00];
            const float b = g[o + i01];
            const float c = g[o + i10];
            const float d = g[o + i11];
            v[j] = fmaf(a, w00, fmaf(b, w01, fmaf(c, w10, d * w11)));
        }
        float4 r;
        r.x = v[0]; r.y = v[1]; r.z = v[2]; r.w = v[3];
        reinterpret_cast<float4*>(op)[cg] = r;
    }
}

extern "C" void kernel_launch(void* const* d_in, const int* in_sizes, int n_in,
                              void* d_out, int out_size, void* d_ws, size_t ws_size,
                              hipStream_t stream) {
    const float* ts    = (const float*)d_in[0];
    const float* theta = (const float*)d_in[1];
    const float* g0    = (const float*)d_in[2];
    const float* g1    = (const float*)d_in[3];
    const float* g2    = (const float*)d_in[4];
    const float* g3    = (const float*)d_in[5];
    float* out = (float*)d_out;

    const int S = in_sizes[0];                    // B*N samples
    // one thread per (sample, level); waves hold a single level
    const long long waves   = ((long long)(S + 31) / 32) * 4;
    const long long threads = waves * 32;
    const int block = 256;
    const int grid  = (int)((threads + block - 1) / block);

    curvegrid_kernel<<<grid, block, 0, stream>>>(ts, theta, g0, g1, g2, g3, out, S);
}